// MultiScaleAttention_12515534701250
// MI455X (gfx1250) — compile-verified
//
#include <hip/hip_runtime.h>
#include <hip/hip_bf16.h>
#include <math.h>

typedef float v2f __attribute__((ext_vector_type(2)));
typedef float v4f __attribute__((ext_vector_type(4)));
typedef float v8f __attribute__((ext_vector_type(8)));

namespace {
constexpr int kS = 1024;
constexpr int kD = 64;
constexpr int kBH = 32;                // B*H
constexpr int kKTot = 1920;            // 1024+512+256+128
constexpr int kPVRows = 896;           // 512+256+128
constexpr size_t kOutElems = (size_t)kBH * kS * kD;  // 2,097,152
// workspace offsets (in floats)
constexpr size_t kWsKproj = 0;
constexpr size_t kWsPooledV = (size_t)kBH * kKTot * kD;                 // 3,932,160
constexpr size_t kWsPVup = kWsPooledV + (size_t)kBH * kPVRows * kD;     // 5,767,168
}  // namespace

static __device__ __forceinline__ v8f wmma4(v2f a, v2f b, v8f c) {
  // D = A(16x4, f32) * B(4x16, f32) + C(16x16, f32)
  return __builtin_amdgcn_wmma_f32_16x16x4_f32(false, a, false, b, (short)0, c,
                                               false, false);
}

static __device__ __forceinline__ void decode_tile(int t, int& si, int& jt) {
  if (t < 64)       { si = 0; jt = t; }
  else if (t < 96)  { si = 1; jt = t - 64; }
  else if (t < 112) { si = 2; jt = t - 96; }
  else              { si = 3; jt = t - 112; }
}

// interp1d(mode='linear', align_corners=False) at output index k, upsampling
// from length Kn to 1024.  invs = Kn/1024.
static __device__ __forceinline__ float lerp_prob(const float* pr, int k,
                                                  int Kn, float invs) {
  float src = ((float)k + 0.5f) * invs - 0.5f;
  src = fminf(fmaxf(src, 0.0f), (float)(Kn - 1));
  const int i0 = (int)src;
  const int i1 = (i0 + 1 < Kn) ? i0 + 1 : Kn - 1;
  const float w = src - (float)i0;
  return pr[i0] * (1.0f - w) + pr[i1] * w;
}

static __device__ __forceinline__ float pup(const float* pr, int k, int Kn,
                                            float invs) {
  if (Kn == kS) return pr[k];
  return lerp_prob(pr, k, Kn, invs);
}

// ---------------------------------------------------------------------------
// Kernel 1: avg-pool K/V per scale; project pooled keys: pk @ W[i]^T + b[i].
// grid = kBH*120 blocks, 128 threads (4 waves; wave w owns output cols w*16..)
// ---------------------------------------------------------------------------
__global__ void msattn_pool_project(const float* __restrict__ key,
                                    const float* __restrict__ val,
                                    const float* __restrict__ W,
                                    const float* __restrict__ bias,
                                    float* __restrict__ kproj,
                                    float* __restrict__ pooledv) {
  __shared__ float pk[16 * kD];
  const int bh = blockIdx.x / 120;
  int si, jt;
  decode_tile(blockIdx.x % 120, si, jt);
  const int s = 1 << si;
  const int offs[4] = {0, 1024, 1536, 1792};
  const int pvo[4] = {0, 0, 512, 768};
  const float inv_s = 1.0f / (float)s;
  const int tid = threadIdx.x;

  for (int e = tid; e < 16 * kD; e += 128) {
    const int j = e >> 6;
    const int d = e & 63;
    const int row0 = (jt * 16 + j) * s;
    const float* kb = key + ((size_t)bh * kS + row0) * kD + d;
    const float* vb = val + ((size_t)bh * kS + row0) * kD + d;
    float ak = 0.f, av = 0.f;
    for (int r = 0; r < s; ++r) {
      ak += kb[r * kD];
      if (si > 0) av += vb[r * kD];
    }
    pk[e] = ak * inv_s;
    if (si > 0)
      pooledv[((size_t)bh * kPVRows + pvo[si] + jt * 16 + j) * kD + d] =
          av * inv_s;
  }
  __syncthreads();

  const int wave = tid >> 5, lane = tid & 31, half = lane >> 4, l16 = lane & 15;
  const int ebase = wave * 16;
  v8f c;
  const float bv = bias[si * kD + ebase + l16];
#pragma unroll
  for (int r = 0; r < 8; ++r) c[r] = bv;
  const float* wrow = W + (size_t)si * kD * kD + (size_t)(ebase + l16) * kD;
#pragma unroll
  for (int j = 0; j < 16; ++j) {
    const int kd = j * 4 + half * 2;
    v2f a;  a.x = pk[l16 * kD + kd];  a.y = pk[l16 * kD + kd + 1];
    v2f b2; b2.x = wrow[kd];          b2.y = wrow[kd + 1];
    c = wmma4(a, b2, c);
  }
  float* kp = kproj + ((size_t)bh * kKTot + offs[si] + jt * 16) * kD;
#pragma unroll
  for (int r = 0; r < 8; ++r) {
    const int m = r + half * 8;
    kp[(size_t)m * kD + ebase + l16] = c[r];
  }
}

// ---------------------------------------------------------------------------
// Kernel 2: upsample pooled V (scales 2,4,8) back to length S.
// idx = ((si*32 + bh)*1024 + k)*64 + d ; grid = 24576 x 256 (exact)
// ---------------------------------------------------------------------------
__global__ void msattn_upsample_v(const float* __restrict__ pooledv,
                                  float* __restrict__ pvup) {
  const int idx = blockIdx.x * 256 + threadIdx.x;
  const int d = idx & 63;
  const int k = (idx >> 6) & 1023;
  const int bh = (idx >> 16) & 31;
  const int si = idx >> 21;  // 0..2 -> scale 2,4,8
  const int Kn = 512 >> si;
  const float invs = 1.0f / (float)(2 << si);
  const int pvo[3] = {0, 512, 768};
  float src = ((float)k + 0.5f) * invs - 0.5f;
  src = fminf(fmaxf(src, 0.0f), (float)(Kn - 1));
  const int i0 = (int)src;
  const int i1 = (i0 + 1 < Kn) ? i0 + 1 : Kn - 1;
  const float w = src - (float)i0;
  const float* base = pooledv + ((size_t)bh * kPVRows + pvo[si]) * kD;
  pvup[idx] =
      base[(size_t)i0 * kD + d] * (1.0f - w) + base[(size_t)i1 * kD + d] * w;
}

// ---------------------------------------------------------------------------
// Kernel 3: fused attention. One block per (bh, 16-query tile), 256 threads,
// dynamic LDS = 16*1920 floats of per-scale probabilities.
// ---------------------------------------------------------------------------
__global__ void msattn_attention(const float* __restrict__ query,
                                 const float* __restrict__ val,
                                 const float* __restrict__ kproj,
                                 const float* __restrict__ pvup,
                                 float* __restrict__ outp,
                                 float* __restrict__ attnp) {
  extern __shared__ float p[];  // [16 rows][1920 keys]
  const int bh = blockIdx.x >> 6;
  const int q0 = (blockIdx.x & 63) * 16;
  const int tid = threadIdx.x;
  const int wave = tid >> 5, lane = tid & 31, half = lane >> 4, l16 = lane & 15;
  const int offs[4] = {0, 1024, 1536, 1792};

  // ---- Phase 1: scores = (q @ Kproj^T) / 8 for all 120 16x16 tiles ----
  const float* qp = query + ((size_t)bh * kS + q0 + l16) * kD;
  for (int t = wave; t < 120; t += 8) {  // 15 tiles per wave, uniform
    int si, kt;
    decode_tile(t, si, kt);
    const float* kp =
        kproj + ((size_t)bh * kKTot + offs[si] + kt * 16 + l16) * kD;
    v8f c = {};
#pragma unroll
    for (int j = 0; j < 16; ++j) {
      const int kd = j * 4 + half * 2;
      v2f a;  a.x = qp[kd];  a.y = qp[kd + 1];
      v2f b2; b2.x = kp[kd]; b2.y = kp[kd + 1];
      c = wmma4(a, b2, c);
    }
    const int kbase = offs[si] + kt * 16;
#pragma unroll
    for (int r = 0; r < 8; ++r) {
      const int m = r + half * 8;
      p[m * kKTot + kbase + l16] = c[r] * 0.125f;
    }
  }
  __syncthreads();

  // ---- Phase 2: softmax per (row, scale); 64 tasks, 8 per wave ----
  for (int ti = 0; ti < 8; ++ti) {
    const int task = wave * 8 + ti;
    const int row = task & 15;
    const int si = task >> 4;
    const int Kn = kS >> si;
    float* pr = p + row * kKTot + offs[si];
    float m = -3.4e38f;
    for (int k = lane; k < Kn; k += 32) m = fmaxf(m, pr[k]);
    for (int o = 16; o > 0; o >>= 1) m = fmaxf(m, __shfl_xor(m, o, 32));
    float sum = 0.f;
    for (int k = lane; k < Kn; k += 32) {
      const float e = expf(pr[k] - m);
      pr[k] = e;
      sum += e;
    }
    for (int o = 16; o > 0; o >>= 1) sum += __shfl_xor(sum, o, 32);
    const float inv = 1.0f / sum;
    for (int k = lane; k < Kn; k += 32) pr[k] *= inv;
  }
  __syncthreads();

  // ---- Phase 3: waves 0-3: output GEMM; waves 4-7: attention mean+store ----
  if (wave < 4) {
    const int dbase = wave * 16;
    v8f c = {};
    for (int si = 0; si < 4; ++si) {
      const int Kn = kS >> si;
      const float invs = 1.0f / (float)(1 << si);
      const float* pvb = (si == 0)
                             ? (val + (size_t)bh * kS * kD)
                             : (pvup + ((size_t)(si - 1) * kBH + bh) * kS * kD);
      const float* pr = p + l16 * kKTot + offs[si];
      for (int kt = 0; kt < 64; ++kt) {
#pragma unroll
        for (int j = 0; j < 4; ++j) {
          const int kk = kt * 16 + j * 4 + half * 2;
          v2f a;
          a.x = pup(pr, kk, Kn, invs);
          a.y = pup(pr, kk + 1, Kn, invs);
          v2f b2;
          b2.x = pvb[(size_t)kk * kD + dbase + l16];
          b2.y = pvb[(size_t)(kk + 1) * kD + dbase + l16];
          c = wmma4(a, b2, c);
        }
      }
    }
    float* op = outp + ((size_t)bh * kS + q0) * kD;
#pragma unroll
    for (int r = 0; r < 8; ++r) {
      const int m = r + half * 8;
      // write-once result: non-temporal store
      __builtin_nontemporal_store(c[r] * 0.25f,
                                  op + (size_t)m * kD + dbase + l16);
    }
  } else {
    const int rw = wave - 4;  // rows rw*4 .. rw*4+3
    float* ap = attnp + ((size_t)bh * kS + q0) * kS;
    for (int rr = 0; rr < 4; ++rr) {
      const int row = rw * 4 + rr;
      const float* pr = p + row * kKTot;
      // 4 consecutive k per lane -> 128-bit non-temporal stores on the
      // dominant (134 MB, write-once) output stream; keeps L2 for Kproj/PVup.
      for (int k0 = lane * 4; k0 < kS; k0 += 128) {
        v4f acc;
#pragma unroll
        for (int u = 0; u < 4; ++u) {
          const int k = k0 + u;
          float a = pr[k];  // scale 1
          a += lerp_prob(pr + 1024, k, 512, 0.5f);
          a += lerp_prob(pr + 1536, k, 256, 0.25f);
          a += lerp_prob(pr + 1792, k, 128, 0.125f);
          acc[u] = a * 0.25f;
        }
        __builtin_nontemporal_store(acc, (v4f*)(ap + (size_t)row * kS + k0));
      }
    }
  }
}

extern "C" void kernel_launch(void* const* d_in, const int* in_sizes, int n_in,
                              void* d_out, int out_size, void* d_ws,
                              size_t ws_size, hipStream_t stream) {
  const float* query = (const float*)d_in[0];
  const float* key = (const float*)d_in[1];
  const float* value = (const float*)d_in[2];
  const float* W = (const float*)d_in[3];
  const float* bias = (const float*)d_in[4];
  float* out = (float*)d_out;
  float* attn = out + kOutElems;
  float* ws = (float*)d_ws;
  float* kproj = ws + kWsKproj;
  float* pooledv = ws + kWsPooledV;
  float* pvupb = ws + kWsPVup;

  // 120 KB dynamic LDS for the attention kernel (320 KB/WGP available).
  hipFuncSetAttribute(reinterpret_cast<const void*>(msattn_attention),
                      hipFuncAttributeMaxDynamicSharedMemorySize,
                      16 * kKTot * (int)sizeof(float));

  msattn_pool_project<<<kBH * 120, 128, 0, stream>>>(key, value, W, bias,
                                                     kproj, pooledv);
  msattn_upsample_v<<<(3 * kBH * kS * kD) / 256, 256, 0, stream>>>(pooledv,
                                                                   pvupb);
  msattn_attention<<<kBH * (kS / 16), 256, 16 * kKTot * sizeof(float),
                     stream>>>(query, value, kproj, pvupb, out, attn);
}